// Eprop_fit_11364483465334
// MI455X (gfx1250) — compile-verified
//
#include <hip/hip_runtime.h>
#include <hip/hip_bf16.h>

#define THRESHOLD  0.6f
#define DAMPENING  0.3f
#define NREF       5
#define DECAY      0.8f
#define ELIG_DECAY 0.8f
#define REG_COEF   300.0f

#define B_    4
#define T_    300
#define N_    200
#define NIN_  100
#define K_    10
#define BT_   (B_ * T_)

// Column-padded staging dims (multiples of 32 so each wave can own a 32-wide
// M-strip with unconditional, in-bounds fragment loads; pad is zero-filled).
#define N_P   224
#define NIN_P 128
#define K_P   16

// Workspace layout (floats)
#define WS_PRE_IN   0
#define WS_PRE_REC  (WS_PRE_IN + BT_ * NIN_P)
#define WS_PRE_OUT  (WS_PRE_REC + BT_ * N_P)
#define WS_POST     (WS_PRE_OUT + BT_ * N_P)
#define WS_G        (WS_POST + BT_ * N_P)
#define WS_E1P      (WS_G + BT_ * N_P)
#define WS_TOTAL    (WS_E1P + BT_ * K_P)

typedef __attribute__((ext_vector_type(2))) float v2f;
typedef __attribute__((ext_vector_type(8))) float v8f;

// ---------------------------------------------------------------------------
// Zero-fill staging workspace (pad rows/cols must be 0; interiors are
// overwritten by the scan kernels afterwards on the same stream).
// ---------------------------------------------------------------------------
__global__ void zero_ws_kernel(float* __restrict__ ws, int n) {
  int i = blockIdx.x * blockDim.x + threadIdx.x;
  if (i < n) ws[i] = 0.0f;
}

// Stage error1 [BT,10] -> padded [BT,16] (pad cols already zero).
__global__ void stage_e1_kernel(const float* __restrict__ e1,
                                float* __restrict__ e1p) {
  int i = blockIdx.x * blockDim.x + threadIdx.x;
  if (i < BT_ * K_) {
    int row = i / K_;
    int k = i - row * K_;
    e1p[row * K_P + k] = e1[i];
  }
}

// ---------------------------------------------------------------------------
// Scan over the time axis: one thread per (b, j).
// Forward: refractory counter -> post_term; z_{t-1} trace -> pre_w_rec;
//          z_t trace -> pre_w_out.
// Backward: L[b,t,j] = error1[b,t,:].w_out[j,:];  Lbar[t] = L[t]+d*Lbar[t+1];
//           g = post_term * Lbar.
// Staged arrays are time-major [BT, padded-cols].
// ---------------------------------------------------------------------------
__global__ void scan_z_kernel(const float* __restrict__ v,
                              const float* __restrict__ z,
                              const float* __restrict__ error1,
                              const float* __restrict__ w_out,
                              float* __restrict__ pre_w_rec,
                              float* __restrict__ pre_w_out,
                              float* __restrict__ post_term,
                              float* __restrict__ g) {
  int idx = blockIdx.x * blockDim.x + threadIdx.x;
  if (idx >= B_ * N_) return;
  int b = idx / N_;
  int j = idx - b * N_;

  float wrow[K_];
#pragma unroll
  for (int k = 0; k < K_; ++k) wrow[k] = w_out[j * K_ + k];

  float tr_prev = 0.f;   // exp_convolve(z_{t-1})
  float tr_cur  = 0.f;   // exp_convolve(z_t)
  float zprev   = 0.f;
  int   ref     = 0;     // refractory count at current t

  const int base = b * T_;
  for (int t = 0; t < T_; ++t) {
    int rin  = (base + t) * N_ + j;    // input arrays [BT, N]
    int rout = (base + t) * N_P + j;   // staged arrays [BT, N_P]
    float zt = z[rin];
    float vt = v[rin];
    float vs  = (vt - THRESHOLD) * (1.0f / THRESHOLD);
    float psi = DAMPENING * fmaxf(1.0f - fabsf(vs), 0.0f) * (1.0f / THRESHOLD);
    float post = (ref > 0) ? 0.0f : psi;
    post_term[rout] = post;

    tr_prev = DECAY * tr_prev + zprev;
    pre_w_rec[rout] = tr_prev;
    tr_cur = DECAY * tr_cur + zt;
    pre_w_out[rout] = tr_cur;

    ref   = (zt > 0.0f) ? (NREF - 1) : ((ref > 0) ? ref - 1 : 0);
    zprev = zt;
  }

  float Lbar = 0.f;
  for (int t = T_ - 1; t >= 0; --t) {
    const float* e = &error1[(base + t) * K_];
    float L = 0.f;
#pragma unroll
    for (int k = 0; k < K_; ++k) L += e[k] * wrow[k];
    Lbar = L + ELIG_DECAY * Lbar;
    int rout = (base + t) * N_P + j;
    g[rout] = post_term[rout] * Lbar;
  }
}

// Input-trace scan: one thread per (b, i).
__global__ void scan_x_kernel(const float* __restrict__ x,
                              float* __restrict__ pre_w_in) {
  int idx = blockIdx.x * blockDim.x + threadIdx.x;
  if (idx >= B_ * NIN_) return;
  int b = idx / NIN_;
  int i = idx - b * NIN_;
  float tr = 0.f;
  for (int t = 0; t < T_; ++t) {
    tr = DECAY * tr + x[(b * T_ + t) * NIN_ + i];
    pre_w_in[(b * T_ + t) * NIN_P + i] = tr;
  }
}

// ---------------------------------------------------------------------------
// WMMA helpers (f32 full-precision path: V_WMMA_F32_16X16X4_F32)
// ---------------------------------------------------------------------------
__device__ __forceinline__ v8f wmma_f32(v2f a, v2f b, v8f c) {
  // 8 args: (neg_a, A, neg_b, B, c_mod, C, reuse_a, reuse_b)
  return __builtin_amdgcn_wmma_f32_16x16x4_f32(false, a, false, b,
                                               (short)0, c, false, false);
}

// Unconditional 2-element fragment load from a padded [BT, ncols] array.
// Element r (r=0,1) holds K = k0 + kh + r, column = col. Always in-bounds.
__device__ __forceinline__ v2f frag_ld(const float* __restrict__ p, int k0,
                                       int kh, int ncols, int col) {
  const float* q = p + (k0 + kh) * ncols + col;
  v2f f;
  f.x = q[0];
  f.y = q[ncols];
  return f;
}

// ---------------------------------------------------------------------------
// GEMM: dW[i,j] = sum_bt pre[bt,i]*g[bt,j]  (+ REG * err2[j] * avg)
// Each wave computes a 32x16 output strip (2 M-tiles) so the two B fragments
// (g, post) are reused across both A fragments: 8 loads -> 4 WMMAs per k-step.
// ---------------------------------------------------------------------------
__global__ void gemm_in_kernel(const float* __restrict__ pre_in,
                               const float* __restrict__ g,
                               const float* __restrict__ post,
                               const float* __restrict__ error2,
                               float* __restrict__ out_in) {
  const int TG = NIN_P / 32;        // 4 strip groups (rows 0..127)
  const int TJ = (N_ + 15) / 16;    // 13 j-tiles
  int wave = threadIdx.x >> 5;
  int lane = threadIdx.x & 31;
  int tile = blockIdx.x * (blockDim.x >> 5) + wave;
  if (tile >= TG * TJ) return;
  int tg = tile / TJ, tj = tile - tg * TJ;

  int m  = lane & 15;
  int kh = (lane >> 4) << 1;       // 0 or 2
  int ia0 = tg * 32 + m;           // A rows, tile 0
  int ia1 = ia0 + 16;              // A rows, tile 1
  int jb  = tj * 16 + m;           // B col (j)

  v8f accg0 = {0.f,0.f,0.f,0.f,0.f,0.f,0.f,0.f};
  v8f accp0 = {0.f,0.f,0.f,0.f,0.f,0.f,0.f,0.f};
  v8f accg1 = {0.f,0.f,0.f,0.f,0.f,0.f,0.f,0.f};
  v8f accp1 = {0.f,0.f,0.f,0.f,0.f,0.f,0.f,0.f};
#pragma unroll 2
  for (int k0 = 0; k0 < BT_; k0 += 4) {
    v2f a0 = frag_ld(pre_in, k0, kh, NIN_P, ia0);
    v2f a1 = frag_ld(pre_in, k0, kh, NIN_P, ia1);
    v2f bg = frag_ld(g,      k0, kh, N_P,   jb);
    v2f bp = frag_ld(post,   k0, kh, N_P,   jb);
    accg0 = wmma_f32(a0, bg, accg0);
    accp0 = wmma_f32(a0, bp, accp0);
    accg1 = wmma_f32(a1, bg, accg1);
    accp1 = wmma_f32(a1, bp, accp1);
  }

  int nj     = tj * 16 + (lane & 15);
  int mbase0 = tg * 32 + ((lane >> 4) << 3);
  if (nj < N_) {
    float e2s = REG_COEF * error2[nj] * (1.0f / (float)BT_);
#pragma unroll
    for (int r = 0; r < 8; ++r) {
      int ii = mbase0 + r;
      if (ii < NIN_) out_in[ii * N_ + nj] = accg0[r] + e2s * accp0[r];
      int ii1 = ii + 16;
      if (ii1 < NIN_) out_in[ii1 * N_ + nj] = accg1[r] + e2s * accp1[r];
    }
  }
}

// dW_rec (200x200) with autapse (diagonal) masking. Same 32x16 strip scheme.
__global__ void gemm_rec_kernel(const float* __restrict__ pre_rec,
                                const float* __restrict__ g,
                                const float* __restrict__ post,
                                const float* __restrict__ error2,
                                float* __restrict__ out_rec) {
  const int TG = N_P / 32;          // 7 strip groups (rows 0..223)
  const int TJ = (N_ + 15) / 16;    // 13 j-tiles
  int wave = threadIdx.x >> 5;
  int lane = threadIdx.x & 31;
  int tile = blockIdx.x * (blockDim.x >> 5) + wave;
  if (tile >= TG * TJ) return;
  int tg = tile / TJ, tj = tile - tg * TJ;

  int m  = lane & 15;
  int kh = (lane >> 4) << 1;
  int ia0 = tg * 32 + m;
  int ia1 = ia0 + 16;
  int jb  = tj * 16 + m;

  v8f accg0 = {0.f,0.f,0.f,0.f,0.f,0.f,0.f,0.f};
  v8f accp0 = {0.f,0.f,0.f,0.f,0.f,0.f,0.f,0.f};
  v8f accg1 = {0.f,0.f,0.f,0.f,0.f,0.f,0.f,0.f};
  v8f accp1 = {0.f,0.f,0.f,0.f,0.f,0.f,0.f,0.f};
#pragma unroll 2
  for (int k0 = 0; k0 < BT_; k0 += 4) {
    v2f a0 = frag_ld(pre_rec, k0, kh, N_P, ia0);
    v2f a1 = frag_ld(pre_rec, k0, kh, N_P, ia1);
    v2f bg = frag_ld(g,       k0, kh, N_P, jb);
    v2f bp = frag_ld(post,    k0, kh, N_P, jb);
    accg0 = wmma_f32(a0, bg, accg0);
    accp0 = wmma_f32(a0, bp, accp0);
    accg1 = wmma_f32(a1, bg, accg1);
    accp1 = wmma_f32(a1, bp, accp1);
  }

  int nj     = tj * 16 + (lane & 15);
  int mbase0 = tg * 32 + ((lane >> 4) << 3);
  if (nj < N_) {
    float e2s = REG_COEF * error2[nj] * (1.0f / (float)BT_);
#pragma unroll
    for (int r = 0; r < 8; ++r) {
      int ii = mbase0 + r;
      if (ii < N_) {
        float val = accg0[r] + e2s * accp0[r];
        out_rec[ii * N_ + nj] = (ii == nj) ? 0.0f : val;
      }
      int ii1 = ii + 16;
      if (ii1 < N_) {
        float val = accg1[r] + e2s * accp1[r];
        out_rec[ii1 * N_ + nj] = (ii1 == nj) ? 0.0f : val;
      }
    }
  }
}

// dW_out[j,k] = sum_bt pre_out[bt,j]*error1[bt,k]  (200x10, k padded to 16).
// 32x16 strip per wave; error1 fragment reused across both A tiles.
__global__ void gemm_out_kernel(const float* __restrict__ pre_out,
                                const float* __restrict__ e1p,
                                float* __restrict__ out_out) {
  const int TG = N_P / 32;  // 7 strip groups
  int wave = threadIdx.x >> 5;
  int lane = threadIdx.x & 31;
  int tile = blockIdx.x * (blockDim.x >> 5) + wave;
  if (tile >= TG) return;

  int m  = lane & 15;
  int kh = (lane >> 4) << 1;
  int ja0 = tile * 32 + m;   // A rows (neuron j), tile 0
  int ja1 = ja0 + 16;        // tile 1

  v8f acc0 = {0.f,0.f,0.f,0.f,0.f,0.f,0.f,0.f};
  v8f acc1 = {0.f,0.f,0.f,0.f,0.f,0.f,0.f,0.f};
#pragma unroll 2
  for (int k0 = 0; k0 < BT_; k0 += 4) {
    v2f a0 = frag_ld(pre_out, k0, kh, N_P, ja0);
    v2f a1 = frag_ld(pre_out, k0, kh, N_P, ja1);
    v2f bk = frag_ld(e1p,     k0, kh, K_P, m);
    acc0 = wmma_f32(a0, bk, acc0);
    acc1 = wmma_f32(a1, bk, acc1);
  }

  int nk     = lane & 15;
  int mbase0 = tile * 32 + ((lane >> 4) << 3);
  if (nk < K_) {
#pragma unroll
    for (int r = 0; r < 8; ++r) {
      int jj = mbase0 + r;
      if (jj < N_) out_out[jj * K_ + nk] = acc0[r];
      int jj1 = jj + 16;
      if (jj1 < N_) out_out[jj1 * K_ + nk] = acc1[r];
    }
  }
}

// ---------------------------------------------------------------------------
extern "C" void kernel_launch(void* const* d_in, const int* in_sizes, int n_in,
                              void* d_out, int out_size, void* d_ws,
                              size_t ws_size, hipStream_t stream) {
  (void)in_sizes; (void)n_in; (void)out_size; (void)ws_size;
  const float* v      = (const float*)d_in[0];  // [B,T,N]
  const float* z      = (const float*)d_in[1];  // [B,T,N]
  const float* x      = (const float*)d_in[2];  // [B,T,NIN]
  const float* error1 = (const float*)d_in[3];  // [B,T,K]
  const float* error2 = (const float*)d_in[4];  // [N]
  const float* w_out  = (const float*)d_in[5];  // [N,K]

  float* out     = (float*)d_out;
  float* out_in  = out;                  // [NIN, N]
  float* out_rec = out_in + NIN_ * N_;   // [N, N]
  float* out_out = out_rec + N_ * N_;    // [N, K]

  float* ws        = (float*)d_ws;
  float* pre_w_in  = ws + WS_PRE_IN;   // [BT, NIN_P]
  float* pre_w_rec = ws + WS_PRE_REC;  // [BT, N_P]
  float* pre_w_out = ws + WS_PRE_OUT;  // [BT, N_P]
  float* post_term = ws + WS_POST;     // [BT, N_P]
  float* g         = ws + WS_G;        // [BT, N_P]
  float* e1p       = ws + WS_E1P;      // [BT, K_P]

  // 1) Zero the staging region (pad rows/cols must be zero).
  zero_ws_kernel<<<(WS_TOTAL + 255) / 256, 256, 0, stream>>>(ws, WS_TOTAL);

  // 2) Stage traces / learning signals.
  stage_e1_kernel<<<(BT_ * K_ + 255) / 256, 256, 0, stream>>>(error1, e1p);
  scan_z_kernel<<<(B_ * N_ + 127) / 128, 128, 0, stream>>>(
      v, z, error1, w_out, pre_w_rec, pre_w_out, post_term, g);
  scan_x_kernel<<<(B_ * NIN_ + 127) / 128, 128, 0, stream>>>(x, pre_w_in);

  // 3) WMMA GEMMs: 8 waves (one 32x16 strip each) per 256-thread block.
  const int WPB = 8;
  const int strips_in  = (NIN_P / 32) * ((N_ + 15) / 16);  // 52
  const int strips_rec = (N_P / 32) * ((N_ + 15) / 16);    // 91
  const int strips_out = (N_P / 32);                       // 7

  gemm_in_kernel<<<(strips_in + WPB - 1) / WPB, 32 * WPB, 0, stream>>>(
      pre_w_in, g, post_term, error2, out_in);
  gemm_rec_kernel<<<(strips_rec + WPB - 1) / WPB, 32 * WPB, 0, stream>>>(
      pre_w_rec, g, post_term, error2, out_rec);
  gemm_out_kernel<<<(strips_out + WPB - 1) / WPB, 32 * WPB, 0, stream>>>(
      pre_w_out, e1p, out_out);
}